// S2Convolution_42691974922375
// MI455X (gfx1250) — compile-verified
//
#include <hip/hip_runtime.h>

// ============================================================================
// S2 spherical convolution (s2cnn first layer) for MI455X / gfx1250.
//
// Pipeline (all fp32, matrix stages on V_WMMA_F32_16X16X4_F32, wave32):
//   0. setup_tables : Wigner-d / quadrature / DFT twiddles (fp64 on device)
//   1. stage_xh     : per-(b,f) alpha-DFT of x (x tile staged to LDS with
//                     GLOBAL_LOAD_ASYNC_TO_LDS_B128, ASYNCcnt-tracked)
//   2. stage_fx     : beta quadrature -> Fx[s,b,i]           (complex)
//   3. stage_fy     : kernel spherical transform Fy[s,i,o]   (complex, + -Im plane)
//   4. per o-chunk (4 x 32 outputs):
//        zero_z     : clear Z (l < lmin slots must be 0)
//        stage_z    : WMMA  Z[mn][l][c] = Fx_block x conj(Fy)_block
//        stage_t    : WMMA  T[j][mn][c] = D[j,l;mn] x Z[mn][l][c]
//        stage_syn  : WMMA  dual 32x31 DFT -> f[b,o,j,a,g]
//                     (T tile fetched by the Tensor Data Mover: 2D D#,
//                      tile 16x961, stride 256 elems, TENSORcnt-tracked;
//                      250KB dynamic LDS per WG, legal on 320KB WGP LDS)
//
// Workspace requirement: 32,730,496 floats = 130,921,984 bytes (~125 MiB)
// -> entire inter-stage working set is L2-resident (192 MB). HBM traffic is
// just x in (8 MB) + out (134 MB) ~ 6 us at 23.3 TB/s; ~20 GFLOP fp32 on the
// WMMA pipes.
// ============================================================================

typedef __attribute__((ext_vector_type(2))) float v2f;
typedef __attribute__((ext_vector_type(8))) float v8f;
typedef __attribute__((ext_vector_type(4))) unsigned int u32x4;
typedef __attribute__((ext_vector_type(8))) int i32x8;
typedef __attribute__((ext_vector_type(4))) int i32x4;

// ---------------- workspace layout (float offsets) ----------------
#define OFF_WD    0u          /* [64][256]           */
#define OFF_EINR  16384u      /* [31][64]            */
#define OFF_EINI  18368u
#define OFF_YKR   20352u      /* [256][24]           */
#define OFF_YKI   26496u
#define OFF_DT    32640u      /* [961][32 j][16 l]   */
#define OFF_CA    524672u     /* [32 a][32 m] cos    */
#define OFF_SA    525696u     /* sin                 */
#define OFF_SN    526720u     /* -sin                */
#define OFF_XHR   527744u     /* [31][8][64][64]     */
#define OFF_XHI   1543552u
#define OFF_FXR   2559360u    /* [2048 rows][64 i]   */
#define OFF_FXI   2690432u
#define OFF_FYR   2821504u    /* [256][64][128]      */
#define OFF_FYI   4918656u
#define OFF_FYN   7015808u    /* -Im(Fy)             */
#define OFF_ZR    9112960u    /* [961][16 l][256 c]  */
#define OFF_ZI    13049216u
#define OFF_TR    16985472u   /* [32 j][961][256 c]  */
#define OFF_TI    24857984u

#define NDT   492032
#define NWD   16384
#define NYK   6144
#define NEIN  1984
#define NCS   1024

// ---------------- fp32 WMMA 16x16x4 wrapper ----------------
static __device__ __forceinline__ v8f wmma4(v2f a, v2f b, v8f c) {
  // 8 args: (neg_a, A, neg_b, B, c_mod, C, reuse_a, reuse_b)
  return __builtin_amdgcn_wmma_f32_16x16x4_f32(false, a, false, b, (short)0, c,
                                               false, false);
}

// ---------------- Tensor Data Mover: 2D tile -> LDS ----------------
// D# per CDNA5 ISA 8.3/8.4: group0 = {count|flags, lds_addr, global_addr,
// type=2}; group1 = {data_size=4B, tensor dims (large: no OOB clamp),
// tile_d0 x tile_d1, dim0 stride in elements}. Groups 2/3 zero (2D tensor).
// This toolchain's builtin takes 6 args (extra int32x8 descriptor block).
static __device__ __forceinline__ void tdm_load_2d(unsigned lds_byte_off,
                                                   const float* gaddr,
                                                   unsigned tile_d0,
                                                   unsigned tile_d1,
                                                   unsigned stride_elems) {
  unsigned long long ga = (unsigned long long)(uintptr_t)gaddr;
  u32x4 g0;
  g0[0] = 1u;                                        // count=1, load, no gather
  g0[1] = lds_byte_off;                              // lds_addr [63:32]
  g0[2] = (unsigned)(ga & 0xFFFFFFFFu);              // global_addr[31:0]
  g0[3] = (unsigned)((ga >> 32) & 0x1FFFFFFu) | (2u << 30);  // [56:32] | type=2
  i32x8 g1;
  g1[0] = (int)(2u << 16);                           // data_size = 4 bytes
  g1[1] = (int)(0xFFFFu << 16);                      // tensor_dim0 lo16 (1M-1)
  g1[2] = (int)(0x000Fu | (0xFFFFu << 16));          // td0 hi | tensor_dim1 lo16
  g1[3] = (int)(0x000Fu | ((tile_d0 & 0xFFFFu) << 16)); // td1 hi | tile_dim0
  g1[4] = (int)(tile_d1 & 0xFFFFu);                  // tile_dim1 | tile_dim2=0
  g1[5] = (int)stride_elems;                         // dim0_stride[31:0]
  g1[6] = 0;                                         // stride hi | dim1_stride
  g1[7] = 0;
  i32x4 z4 = {0, 0, 0, 0};
  i32x8 z8 = {0, 0, 0, 0, 0, 0, 0, 0};
  __builtin_amdgcn_tensor_load_to_lds(g0, g1, z4, z4, z8, 0);
}

// ---------------- fp64 Wigner-d helpers ----------------
static __device__ double dfact(int n) {
  double r = 1.0;
  for (int i = 2; i <= n; ++i) r *= (double)i;
  return r;
}
static __device__ double dipow(double b, int e) {
  double r = 1.0;
  for (int i = 0; i < e; ++i) r *= b;
  return r;
}
static __device__ double wig_d(int l, int m, int n, double beta) {
  double pref = sqrt(dfact(l + m) * dfact(l - m) * dfact(l + n) * dfact(l - n));
  double cb = cos(0.5 * beta), sb = sin(0.5 * beta);
  int s0 = (n - m) > 0 ? (n - m) : 0;
  int s1 = (l + n) < (l - m) ? (l + n) : (l - m);
  double out = 0.0;
  for (int s = s0; s <= s1; ++s) {
    double coef = (((m - n + s) & 1) ? -1.0 : 1.0) /
                  (dfact(l + n - s) * dfact(s) * dfact(m - n + s) * dfact(l - m - s));
    out += coef * dipow(cb, 2 * l + n - m - 2 * s) * dipow(sb, m - n + 2 * s);
  }
  return pref * out;
}

// ---------------- 0: constant tables ----------------
__global__ void setup_tables(float* ws) {
  int t = blockIdx.x * blockDim.x + threadIdx.x;
  const double PI = 3.14159265358979323846;
  if (t < NDT) {  // D_OUT, layout [mn][j][l]
    int l = t & 15, j = (t >> 4) & 31, mn = t >> 9;
    int np = mn % 31, mp = mn / 31;
    int m = mp - 15, n = np - 15;
    int am = m < 0 ? -m : m, an = n < 0 ? -n : n;
    int lmin = am > an ? am : an;
    float v = 0.f;
    if (l >= lmin) v = (float)((double)(2 * l + 1) * wig_d(l, m, n, (j + 0.5) * PI / 32.0));
    ws[OFF_DT + (size_t)(mn * 32 + j) * 16 + l] = v;
    return;
  }
  t -= NDT;
  if (t < NWD) {  // WD_IN [j][s] incl. DH quadrature weight * alpha cell
    int s = t & 255, j = t >> 8;
    int l = 0; while ((l + 1) * (l + 1) <= s) ++l;
    int m = s - l * l - l;
    double beta = (j + 0.5) * PI / 64.0;
    double acc = 0.0;
    for (int k = 0; k < 32; ++k)
      acc += sin((double)(2 * j + 1) * (double)(2 * k + 1) * PI / 128.0) / (double)(2 * k + 1);
    double w = (2.0 / 32.0) * sin(beta) * acc;
    ws[OFF_WD + j * 256 + s] = (float)(wig_d(l, m, 0, beta) * w * (2.0 * PI / 64.0));
    return;
  }
  t -= NWD;
  if (t < NYK) {  // YK [s][k]
    int k = t % 24, s = t / 24;
    int l = 0; while ((l + 1) * (l + 1) <= s) ++l;
    int m = s - l * l - l;
    double gb = (double)((k >> 3) + 1) * PI / 24.0;
    double ga = (double)(k & 7) * PI / 4.0;
    double d = wig_d(l, m, 0, gb);
    ws[OFF_YKR + s * 24 + k] = (float)(d * cos((double)m * ga));
    ws[OFF_YKI + s * 24 + k] = (float)(-d * sin((double)m * ga));
    return;
  }
  t -= NYK;
  if (t < NEIN) {  // e^{-i m a 2pi/64}, [mp][a]
    int mp = t / 64, a = t % 64;
    double ang = -(2.0 * PI / 64.0) * (double)(mp - 15) * (double)a;
    ws[OFF_EINR + t] = (float)cos(ang);
    ws[OFF_EINI + t] = (float)sin(ang);
    return;
  }
  t -= NEIN;
  if (t < NCS) {  // synthesis twiddles [a][m], col 31 = K-pad zero
    int a = t >> 5, col = t & 31;
    float c = 0.f, sv = 0.f;
    if (col < 31) {
      double ang = (2.0 * PI / 32.0) * (double)(col - 15) * (double)a;
      c = (float)cos(ang);
      sv = (float)sin(ang);
    }
    ws[OFF_CA + t] = c;
    ws[OFF_SA + t] = sv;
    ws[OFF_SN + t] = -sv;
  }
}

// ---------------- 1: alpha DFT at 31 bins, Xh[mp][b][f][j] ----------------
__global__ void stage_xh(float* ws, const float* x) {
  __shared__ float xt[4096];
  __shared__ float er[NEIN], ei[NEIN];
  int bf = blockIdx.x;            // b*64 + f
  int tid = threadIdx.x;

  // async-copy the contiguous 16KB x tile into LDS (b128 per lane)
  const float* xblk = x + (size_t)bf * 4096;
  unsigned ldsbase = (unsigned)(uintptr_t)&xt[0];  // LDS offset = addr[31:0]
  for (int i = tid; i < 1024; i += 256) {
    unsigned ldsoff = ldsbase + (unsigned)i * 16u;
    int goff = i * 16;
    asm volatile("global_load_async_to_lds_b128 %0, %1, %2 offset:0"
                 :: "v"(ldsoff), "v"(goff), "s"(xblk) : "memory");
  }
  for (int i = tid; i < NEIN; i += 256) { er[i] = ws[OFF_EINR + i]; ei[i] = ws[OFF_EINI + i]; }
  asm volatile("s_wait_asynccnt 0" ::: "memory");
  __syncthreads();

  int b = bf >> 6, f = bf & 63;
  for (int p = tid; p < 31 * 64; p += 256) {
    int mp = p >> 6, j = p & 63;
    float ar = 0.f, ai = 0.f;
    for (int a = 0; a < 64; ++a) {
      float xv = xt[j * 64 + a];
      ar += xv * er[mp * 64 + a];
      ai += xv * ei[mp * 64 + a];
    }
    size_t o = ((size_t)(mp * 8 + b) * 64 + f) * 64 + j;
    ws[OFF_XHR + o] = ar;
    ws[OFF_XHI + o] = ai;
  }
}

// ---------------- 2: beta quadrature, Fx rows = (s*8+b), cols i --------------
__global__ void stage_fx(float* ws) {
  int s = blockIdx.x;
  int tid = threadIdx.x;          // 512: b*64 + f
  int b = tid >> 6, f = tid & 63;
  int l = 0; while ((l + 1) * (l + 1) <= s) ++l;
  int mp = (s - l * l - l) + 15;
  size_t xb = ((size_t)(mp * 8 + b) * 64 + f) * 64;
  float ar = 0.f, ai = 0.f;
  for (int j = 0; j < 64; ++j) {
    float wd = ws[OFF_WD + j * 256 + s];
    ar += wd * ws[OFF_XHR + xb + j];
    ai += wd * ws[OFF_XHI + xb + j];
  }
  ws[OFF_FXR + (size_t)(s * 8 + b) * 64 + f] = ar;
  ws[OFF_FXI + (size_t)(s * 8 + b) * 64 + f] = ai;
}

// ---------------- 3: kernel spherical transform Fy[s][i][o] ----------------
__global__ void stage_fy(float* ws, const float* ker) {
  const float SCALE = 0.07216878364870323f;  // 1/sqrt(24*64*16^3/32^3)
  int bid = blockIdx.x;           // s*64 + i
  int s = bid >> 6, i = bid & 63;
  int o = threadIdx.x;            // 128
  float ar = 0.f, ai = 0.f;
  for (int k = 0; k < 24; ++k) {
    float kv = ker[(size_t)(i * 128 + o) * 24 + k] * SCALE;
    ar += kv * ws[OFF_YKR + s * 24 + k];
    ai += kv * ws[OFF_YKI + s * 24 + k];
  }
  size_t idx = (size_t)(s * 64 + i) * 128 + o;
  ws[OFF_FYR + idx] = ar;
  ws[OFF_FYI + idx] = ai;
  ws[OFF_FYN + idx] = -ai;
}

// ---------------- 4a: clear Z (l < lmin slots feed zero D rows) --------------
__global__ void zero_z(float* ws) {
  const int count = 2 * 3936256;
  int stride = gridDim.x * blockDim.x;
  for (int i = blockIdx.x * blockDim.x + threadIdx.x; i < count; i += stride)
    ws[OFF_ZR + i] = 0.f;
}

// ---------------- 4b: Z[mn][l][c] = Fx_block x conj(Fy)_block (WMMA) ---------
__global__ void stage_z(float* ws, int ochunk) {
  int wid = (blockIdx.x * blockDim.x + threadIdx.x) >> 5;
  if (wid >= 5712) return;
  int lane = threadIdx.x & 31, laneM = lane & 15, laneH = lane >> 4;
  int acc = 0, l = 0;
  for (l = 0; l < 16; ++l) {
    int cnt = (l + 1) * (2 * l + 1) * 2;
    if (wid < acc + cnt) break;
    acc += cnt;
  }
  int rem = wid - acc;
  int inner = (l + 1) * 2;
  int ni = rem / inner, r2 = rem % inner;
  int mt = r2 >> 1, nt = r2 & 1;

  int localRows = (2 * l + 1) * 8;
  int row = mt * 16 + laneM;
  bool valid = row < localRows;
  size_t growB = (size_t)(l * l * 8 + row) * 64;
  int s2 = l * l + ni;
  int oCol = ochunk * 32 + nt * 16 + laneM;

  const float* fxr = ws + OFF_FXR;
  const float* fxi = ws + OFF_FXI;
  const float* fyr = ws + OFF_FYR;
  const float* fyi = ws + OFF_FYI;
  const float* fyn = ws + OFF_FYN;

  v8f zr = {0.f, 0.f, 0.f, 0.f, 0.f, 0.f, 0.f, 0.f};
  v8f zi = {0.f, 0.f, 0.f, 0.f, 0.f, 0.f, 0.f, 0.f};
  for (int kt = 0; kt < 16; ++kt) {
    int ka = kt * 4 + 2 * laneH;
    v2f ar = {0.f, 0.f}, ai = {0.f, 0.f};
    if (valid) {
      ar.x = fxr[growB + ka]; ar.y = fxr[growB + ka + 1];
      ai.x = fxi[growB + ka]; ai.y = fxi[growB + ka + 1];
    }
    size_t b0 = (size_t)(s2 * 64 + ka) * 128 + oCol;
    v2f byr = {fyr[b0], fyr[b0 + 128]};
    v2f byi = {fyi[b0], fyi[b0 + 128]};
    v2f byn = {fyn[b0], fyn[b0 + 128]};
    // X * conj(Y): re = xr*yr + xi*yi ; im = xi*yr - xr*yi
    zr = wmma4(ar, byr, zr);
    zr = wmma4(ai, byi, zr);
    zi = wmma4(ai, byr, zi);
    zi = wmma4(ar, byn, zi);
  }
  int npv = ni + 15 - l;
  for (int v = 0; v < 8; ++v) {
    int rrow = mt * 16 + v + 8 * laneH;
    if (rrow < localRows) {
      int mi = rrow >> 3, b = rrow & 7;
      int mpv = mi + 15 - l;
      int c = b * 32 + nt * 16 + laneM;
      size_t zidx = ((size_t)(mpv * 31 + npv) * 16 + l) * 256 + c;
      ws[OFF_ZR + zidx] = zr[v];
      ws[OFF_ZI + zidx] = zi[v];
    }
  }
}

// ---------------- 4c: T[j][mn][c] = D[mn] (32x16) x Z[mn] (16x256) (WMMA) ----
__global__ void stage_t(float* ws) {
  int wid = (blockIdx.x * blockDim.x + threadIdx.x) >> 5;
  if (wid >= 30752) return;
  int lane = threadIdx.x & 31, laneM = lane & 15, laneH = lane >> 4;
  int mn = wid >> 5, r = wid & 31, mt = r >> 4, nt = r & 15;

  const float* A = ws + OFF_DT + (size_t)mn * 512;
  const float* Br = ws + OFF_ZR + (size_t)mn * 16 * 256;
  const float* Bi = ws + OFF_ZI + (size_t)mn * 16 * 256;

  v8f tr = {0.f, 0.f, 0.f, 0.f, 0.f, 0.f, 0.f, 0.f};
  v8f ti = {0.f, 0.f, 0.f, 0.f, 0.f, 0.f, 0.f, 0.f};
  for (int kt = 0; kt < 4; ++kt) {
    int ka = kt * 4 + 2 * laneH;
    int arow = (mt * 16 + laneM) * 16;
    v2f a = {A[arow + ka], A[arow + ka + 1]};
    int b0 = ka * 256 + nt * 16 + laneM;
    v2f br = {Br[b0], Br[b0 + 256]};
    v2f bi = {Bi[b0], Bi[b0 + 256]};
    tr = wmma4(a, br, tr);
    ti = wmma4(a, bi, ti);
  }
  for (int v = 0; v < 8; ++v) {
    int j = mt * 16 + v + 8 * laneH;
    int c = nt * 16 + laneM;
    size_t idx = ((size_t)j * 961 + mn) * 256 + c;
    ws[OFF_TR + idx] = tr[v];
    ws[OFF_TI + idx] = ti[v];
  }
}

// ---------------- 4d: dual DFT synthesis per (j, c16) tile (WMMA) -----------
// LDS: T slice (2x15376) | U (2x15872); f-buffer aliases T region.
// T slice is fetched by the Tensor Data Mover (one 2D descriptor per plane).
__global__ void stage_syn(const float* ws, float* out, int ochunk) {
  extern __shared__ float lds[];
  float* sTr = lds;
  float* sTi = lds + 15376;
  float* sUr = lds + 30752;
  float* sUi = lds + 46624;
  float* sF  = lds;  // 16384, reuses T region after phase 2

  int tid = threadIdx.x;
  int lane = tid & 31, laneM = lane & 15, laneH = lane >> 4;
  int wv = tid >> 5;
  int j = blockIdx.x >> 4, ct = blockIdx.x & 15, cbase = ct * 16;

  const float* tre = ws + OFF_TR;
  const float* tim = ws + OFF_TI;
  const float* ca = ws + OFF_CA;
  const float* sa = ws + OFF_SA;
  const float* sn = ws + OFF_SN;

  // phase 1: TDM-load T[j][mn][cbase..+16) into LDS (tile 16 x 961, stride 256)
  if (tid < 32) {  // one wave owns the DMA; TDM ignores EXEC
    size_t gbase = (size_t)j * 961 * 256 + cbase;
    tdm_load_2d(0u,          tre + gbase, 16u, 961u, 256u);
    tdm_load_2d(15376u * 4u, tim + gbase, 16u, 961u, 256u);
    __builtin_amdgcn_s_wait_tensorcnt(0);
  }
  __syncthreads();

  // phase 2: U[a][n][c] = sum_m E[a][m] * T[m][n][c]   (complex * complex)
  for (int tile = wv; tile < 62; tile += 8) {
    int n = tile >> 1, at = tile & 1;
    v8f ur = {0.f, 0.f, 0.f, 0.f, 0.f, 0.f, 0.f, 0.f};
    v8f ui = {0.f, 0.f, 0.f, 0.f, 0.f, 0.f, 0.f, 0.f};
    for (int kt = 0; kt < 8; ++kt) {
      int ka = kt * 4 + 2 * laneH;
      int arow = (at * 16 + laneM) * 32;
      v2f ac = {ca[arow + ka], ca[arow + ka + 1]};
      v2f as_ = {sa[arow + ka], sa[arow + ka + 1]};
      v2f an = {sn[arow + ka], sn[arow + ka + 1]};
      int b0 = (ka * 31 + n) * 16 + laneM;
      int b1 = ((ka + 1) * 31 + n) * 16 + laneM;
      v2f br = {sTr[b0], sTr[b1]};
      v2f bi = {sTi[b0], sTi[b1]};
      ur = wmma4(ac, br, ur);   // Ure = C*Tre - S*Tim
      ur = wmma4(an, bi, ur);
      ui = wmma4(ac, bi, ui);   // Uim = C*Tim + S*Tre
      ui = wmma4(as_, br, ui);
    }
    for (int v = 0; v < 8; ++v) {
      int a = at * 16 + v + 8 * laneH;
      sUr[(a * 31 + n) * 16 + laneM] = ur[v];
      sUi[(a * 31 + n) * 16 + laneM] = ui[v];
    }
  }
  __syncthreads();

  // phase 3: f[a][g][c] = sum_n (cos*Ure - sin*Uim)
  for (int tile = wv; tile < 64; tile += 8) {
    int a = tile >> 1, gt = tile & 1;
    v8f fv = {0.f, 0.f, 0.f, 0.f, 0.f, 0.f, 0.f, 0.f};
    for (int kt = 0; kt < 8; ++kt) {
      int ka = kt * 4 + 2 * laneH;
      int grow = (gt * 16 + laneM) * 32;
      v2f ac = {ca[grow + ka], ca[grow + ka + 1]};
      v2f an = {sn[grow + ka], sn[grow + ka + 1]};
      int b0 = (a * 31 + ka) * 16 + laneM;
      int b1 = (a * 31 + ka + 1) * 16 + laneM;
      v2f br = {sUr[b0], sUr[b1]};
      v2f bi = {sUi[b0], sUi[b1]};
      fv = wmma4(ac, br, fv);
      fv = wmma4(an, bi, fv);
    }
    for (int v = 0; v < 8; ++v) {
      int g = gt * 16 + v + 8 * laneH;
      sF[laneM * 1024 + a * 32 + g] = fv[v];
    }
  }
  __syncthreads();

  // phase 4: coalesced store to out[b][o][j][a][g]
  for (int idx = tid; idx < 16384; idx += 256) {
    int c = idx >> 10, rem = idx & 1023;  // rem = a*32+g
    int b = c >> 5, op = c & 31;
    int o = ochunk * 32 + op;
    out[(((size_t)(b * 128 + o) * 32 + j) << 10) + rem] = sF[idx];
  }
}

// ---------------- host-side launch ----------------
extern "C" void kernel_launch(void* const* d_in, const int* in_sizes, int n_in,
                              void* d_out, int out_size, void* d_ws, size_t ws_size,
                              hipStream_t stream) {
  (void)in_sizes; (void)n_in; (void)out_size; (void)ws_size;
  const float* x = (const float*)d_in[0];     // [8,64,64,64]
  const float* ker = (const float*)d_in[1];   // [64,128,24]
  float* out = (float*)d_out;                 // [8,128,32,32,32]
  float* ws = (float*)d_ws;                   // needs >= 130,921,984 bytes

  setup_tables<<<2022, 256, 0, stream>>>(ws);
  stage_xh<<<512, 256, 0, stream>>>(ws, x);
  stage_fx<<<256, 512, 0, stream>>>(ws);
  stage_fy<<<16384, 128, 0, stream>>>(ws, ker);

  for (int oc = 0; oc < 4; ++oc) {
    zero_z<<<3844, 256, 0, stream>>>(ws);
    stage_z<<<714, 256, 0, stream>>>(ws, oc);
    stage_t<<<3844, 256, 0, stream>>>(ws);
    stage_syn<<<512, 256, 249984, stream>>>(ws, out, oc);
  }
}